// SphericalFourierEncoding_670014898690
// MI455X (gfx1250) — compile-verified
//
#include <hip/hip_runtime.h>

typedef __attribute__((ext_vector_type(2))) float v2f;
typedef __attribute__((ext_vector_type(8))) float v8f;

#define LMAXV   6
#define NRADV   20
#define NELEMV  4
#define NLMV    49          // (LMAX+1)^2
#define CHUNK   128
#define FSTRIDE 96          // [rad 0..19][zero 20..31][Y 32..80][zero 81..95]
#define OUT_PER_BATCH 3920  // NELEM * NRAD * NLM

// ---- compile-time norm table: nlm = sqrt((2l+1)(l-|m|)!/(l+|m|)!), *sqrt(2) for m!=0
constexpr double csqrt_c(double x) {
  double g = x > 1.0 ? x : 1.0;
  for (int i = 0; i < 80; ++i) g = 0.5 * (g + x / g);
  return g;
}
constexpr double cfact_c(int n) {
  double f = 1.0;
  for (int i = 2; i <= n; ++i) f *= (double)i;
  return f;
}
struct NormTab { float v[NLMV]; };
constexpr NormTab make_norm() {
  NormTab t{};
  for (int l = 0; l <= LMAXV; ++l)
    for (int m = -l; m <= l; ++m) {
      int am = m < 0 ? -m : m;
      double nlm = csqrt_c((2.0 * l + 1.0) * cfact_c(l - am) / cfact_c(l + am));
      double c = (m == 0) ? nlm : nlm * csqrt_c(2.0);
      t.v[l * l + m + l] = (float)c;
    }
  return t;
}
__constant__ NormTab NORM = make_norm();

// ---- per-point features into one LDS row (96 floats, zero-padded regions)
__device__ __forceinline__ void compute_features(float x, float y, float z, float* row) {
  float r2 = x * x + y * y + z * z;
  float r = sqrtf(r2);
  float inv = 1.0f / r;
  float ux = x * inv, uy = y * inv, uz = z * inv;

  // radial: sqrt(2/RC) * sin(n*pi*r/RC)/r via Chebyshev recurrence on sin(n t)
  const float t = 0.31415926535897931f * r;  // pi/RC, RC=10
  float s, c;
  __sincosf(t, &s, &c);
  float twoc = 2.0f * c;
  float sprev = 0.0f, scur = s;
  const float kr = 0.44721359549995793f * inv;  // sqrt(2/RC)/r
#pragma unroll
  for (int n = 0; n < NRADV; ++n) {
    row[n] = kr * scur;
    float snext = twoc * scur - sprev;
    sprev = scur;
    scur = snext;
  }
#pragma unroll
  for (int i = NRADV; i < 32; ++i) row[i] = 0.0f;

  // real spherical harmonics, l=0..6, component normalization
  float cs[LMAXV + 1], sn[LMAXV + 1];
  cs[0] = 1.0f; sn[0] = 0.0f;
#pragma unroll
  for (int m = 1; m <= LMAXV; ++m) {
    cs[m] = cs[m - 1] * ux - sn[m - 1] * uy;
    sn[m] = sn[m - 1] * ux + cs[m - 1] * uy;
  }
  float Q[LMAXV + 1][LMAXV + 1];
  Q[0][0] = 1.0f;
#pragma unroll
  for (int m = 1; m <= LMAXV; ++m) Q[m][m] = -(2.0f * m - 1.0f) * Q[m - 1][m - 1];
#pragma unroll
  for (int m = 0; m < LMAXV; ++m) Q[m + 1][m] = (2.0f * m + 1.0f) * uz * Q[m][m];
#pragma unroll
  for (int m = 0; m <= LMAXV; ++m) {
#pragma unroll
    for (int l = m + 2; l <= LMAXV; ++l) {
      Q[l][m] = ((2.0f * l - 1.0f) * uz * Q[l - 1][m] - (float)(l + m - 1) * Q[l - 2][m]) *
                (1.0f / (float)(l - m));
    }
  }
  int idx = 32;
#pragma unroll
  for (int l = 0; l <= LMAXV; ++l) {
#pragma unroll
    for (int mm = -l; mm <= l; ++mm) {
      int am = mm < 0 ? -mm : mm;
      float trig = (mm < 0) ? sn[am] : ((mm > 0) ? cs[am] : 1.0f);
      row[idx++] = NORM.v[l * l + mm + l] * Q[l][am] * trig;
    }
  }
#pragma unroll
  for (int i = 32 + NLMV; i < FSTRIDE; ++i) row[i] = 0.0f;
}

// One block per batch; wave w handles elem w. Segment-sum == GEMM (M=20,N=49,K=pts)
// accumulated with exact-f32 V_WMMA_F32_16X16X4_F32. No global atomics.
__global__ __launch_bounds__(128) void sfe_kernel(const float* __restrict__ coords,
                                                  const int* __restrict__ elem_ids,
                                                  const int* __restrict__ batch_ids,
                                                  float* __restrict__ out, int npts) {
  __shared__ float feat[CHUNK + 1][FSTRIDE];  // row CHUNK = zero dummy row
  __shared__ int   plist[NELEMV][CHUNK + 4];
  __shared__ int   eid[CHUNK];
  __shared__ int   kpad[NELEMV];
  __shared__ int   srange[2];

  const int tid = threadIdx.x;
  const int b = blockIdx.x;

  if (tid == 0) {
    int lo = 0, hi = npts;
    while (lo < hi) { int mid = (lo + hi) >> 1; if (batch_ids[mid] < b) lo = mid + 1; else hi = mid; }
    srange[0] = lo;
    lo = 0; hi = npts;
    while (lo < hi) { int mid = (lo + hi) >> 1; if (batch_ids[mid] < b + 1) lo = mid + 1; else hi = mid; }
    srange[1] = lo;
  }
  if (tid < FSTRIDE) feat[CHUNK][tid] = 0.0f;  // dummy row for K padding
  __syncthreads();

  const int start = srange[0];
  const int end = srange[1];

  const int lane = tid & 31;
  const int wv = tid >> 5;     // elem id this wave owns
  const int half = lane >> 4;  // K-pair selector
  const int col = lane & 15;   // M (A) / N (B,D) column

  v8f acc[2][4];
#pragma unroll
  for (int mt = 0; mt < 2; ++mt)
#pragma unroll
    for (int nt = 0; nt < 4; ++nt)
#pragma unroll
      for (int i = 0; i < 8; ++i) acc[mt][nt][i] = 0.0f;

  for (int cbase = start; cbase < end; cbase += CHUNK) {
    int p = cbase + tid;
    int e = 255;
    if (p < end) {
      float x = coords[3 * p + 0];
      float y = coords[3 * p + 1];
      float z = coords[3 * p + 2];
      compute_features(x, y, z, feat[tid]);
      e = elem_ids[p];
    }
    eid[tid] = e;
    __syncthreads();

    // deterministic per-elem compaction (serial scan, no atomics)
    if (tid < NELEMV) {
      int K = 0;
      for (int i = 0; i < CHUNK; ++i)
        if (eid[i] == tid) plist[tid][K++] = i;
      int Kp = (K + 3) & ~3;
      for (int i = K; i < Kp; ++i) plist[tid][i] = CHUNK;  // zero dummy row
      kpad[tid] = Kp;
    }
    __syncthreads();

    const int Ke = kpad[wv];  // wave-uniform
    for (int k = 0; k < Ke; k += 4) {
      int pa = plist[wv][k + 2 * half];
      int pb = plist[wv][k + 2 * half + 1];
      const float* ra = feat[pa];
      const float* rb = feat[pb];
      // A (16x4, M=rad index n): v0/v1 = K pair; rows 16..31 read zero pad for n>=20
      v2f a0, a1;
      a0.x = ra[col];       a0.y = rb[col];
      a1.x = ra[16 + col];  a1.y = rb[16 + col];
      // B (4x16, N=lm): zero pad covers lm>=49
      v2f bf[4];
#pragma unroll
      for (int nt = 0; nt < 4; ++nt) {
        bf[nt].x = ra[32 + nt * 16 + col];
        bf[nt].y = rb[32 + nt * 16 + col];
      }
#pragma unroll
      for (int nt = 0; nt < 4; ++nt) {
        acc[0][nt] = __builtin_amdgcn_wmma_f32_16x16x4_f32(false, a0, false, bf[nt],
                                                           (short)0, acc[0][nt], false, false);
        acc[1][nt] = __builtin_amdgcn_wmma_f32_16x16x4_f32(false, a1, false, bf[nt],
                                                           (short)0, acc[1][nt], false, false);
      }
    }
    __syncthreads();  // feat rows reused next chunk
  }

  // exclusive stores into final layout: out[b, 80*l^2 + e*20*(2l+1) + n*(2l+1) + mi]
  const int base = b * OUT_PER_BATCH;
#pragma unroll
  for (int mt = 0; mt < 2; ++mt) {
#pragma unroll
    for (int nt = 0; nt < 4; ++nt) {
#pragma unroll
      for (int i = 0; i < 8; ++i) {
        int n = mt * 16 + 8 * half + i;
        int lm = nt * 16 + col;
        if (n < NRADV && lm < NLMV) {
          int l = (lm >= 36) ? 6 : (lm >= 25) ? 5 : (lm >= 16) ? 4
                : (lm >= 9) ? 3 : (lm >= 4) ? 2 : (lm >= 1) ? 1 : 0;
          int mi = lm - l * l;
          int w = 2 * l + 1;
          out[base + 80 * l * l + wv * NRADV * w + n * w + mi] = acc[mt][nt][i];
        }
      }
    }
  }
}

extern "C" void kernel_launch(void* const* d_in, const int* in_sizes, int n_in,
                              void* d_out, int out_size, void* d_ws, size_t ws_size,
                              hipStream_t stream) {
  const float* coords = (const float*)d_in[0];
  const int* elem_ids = (const int*)d_in[1];
  const int* batch_ids = (const int*)d_in[2];
  float* out = (float*)d_out;
  int npts = in_sizes[1];                 // NPTS
  int nbatch = out_size / OUT_PER_BATCH;  // NBATCH
  sfe_kernel<<<dim3(nbatch), dim3(128), 0, stream>>>(coords, elem_ids, batch_ids, out, npts);
}